// mpNet_1486058684581
// MI455X (gfx1250) — compile-verified
//
#include <hip/hip_runtime.h>

// MI455X (gfx1250) hybrid matching-pursuit:
//  - scoring matmul on V_WMMA_F32_16X16X4_F32 (FP32, matrix units)
//  - selected coefficient + residual update in exact FP64 (VALU)
// Residual tile stays resident in LDS across all k iterations.
// v2: fold the Im-term minus sign into register-resident A fragments
//     (arn = -ar), eliminating the third packed W array -> 32 global
//     b64 loads per 64 WMMAs instead of 48.

typedef __attribute__((ext_vector_type(2))) float v2f;
typedef __attribute__((ext_vector_type(8))) float v8f;

#define BB 2048
#define NN 64
#define DD 4096
#define OUT_HALF (BB * NN * 2)   // doubles per output tensor (interleaved complex)

// ---------------------------------------------------------------------------
// Pack W (complex128, row-major [N][D]) into two FP32 arrays laid out in the
// exact per-lane order the WMMA B-operand wants:
//   o = ((ct*16 + s)*32 + lane)*2 + j   with  k = s*4 + j + 2*(lane>=16),
//   c = ct*16 + (lane&15)
// so each lane fetches its v2f fragment with one global float2 load.
// ---------------------------------------------------------------------------
__global__ void pack_W_kernel(const double* __restrict__ W,
                              float* __restrict__ Wr,
                              float* __restrict__ Wi) {
    int o = blockIdx.x * blockDim.x + threadIdx.x;
    if (o >= NN * DD) return;
    int j    = o & 1;
    int lane = (o >> 1) & 31;
    int sct  = o >> 6;            // ct*16 + s
    int s    = sct & 15;
    int ct   = sct >> 4;
    int k    = s * 4 + j + 2 * (lane >> 4);
    int c    = ct * 16 + (lane & 15);
    size_t src = (size_t)(k * DD + c) * 2;
    Wr[o] = (float)W[src];
    Wi[o] = (float)W[src + 1];
}

// ---------------------------------------------------------------------------
// Main kernel: one block = 16 residual rows, 512 threads = 16 waves.
// Each wave sweeps 16 column-tiles (16 cols each) per iteration: 64 WMMAs/tile.
// ---------------------------------------------------------------------------
__global__ __launch_bounds__(512)
void mp_wmma_kernel(const double* __restrict__ x,
                    const double* __restrict__ W,
                    const int*    __restrict__ kptr,
                    double*       __restrict__ out,
                    const float*  __restrict__ Wr,
                    const float*  __restrict__ Wi) {
    // LDS: FP64 residual tile + FP32 shadow for WMMA A-fragments (row-padded
    // to dodge the 256B bank period), per-wave argmax scratch, selections.
    __shared__ double sRe[16][66];
    __shared__ double sIm[16][66];
    __shared__ float  sAr[16][68];
    __shared__ float  sAi[16][68];
    __shared__ float  redV[16][16];   // [wave][row]
    __shared__ int    redC[16][16];
    __shared__ int    sCol[16];
    __shared__ double sSelRe[16], sSelIm[16];

    const int tid  = threadIdx.x;
    const int wave = tid >> 5;
    const int lane = tid & 31;
    const int row0 = blockIdx.x * 16;

    // ---- load x tile -> residual (FP64 + FP32 shadow) ----
    for (int e = tid; e < 16 * NN; e += 512) {
        int r = e >> 6, n = e & 63;
        size_t gi = ((size_t)(row0 + r) * NN + n) * 2;
        double re = x[gi], im = x[gi + 1];
        sRe[r][n] = re;        sIm[r][n] = im;
        sAr[r][n] = (float)re; sAi[r][n] = (float)im;
    }
    __syncthreads();

    const int K    = kptr[0];
    const int m    = lane & 15;        // WMMA row this lane feeds (A operand)
    const int koff = 2 * (lane >> 4);  // K shift for upper half-wave

    for (int iter = 0; iter < K; ++iter) {
        // ---- preload all 16 A-fragments (K=0..63 in steps of 4) ----
        // arn = -ar carries the Im-term minus sign so only two B arrays
        // (Wr, Wi) are ever streamed from memory.
        v2f ar[16], ai[16], arn[16];
#pragma unroll
        for (int s = 0; s < 16; ++s) {
            int k = s * 4 + koff;
            float2 fa = *(const float2*)&sAr[m][k];
            float2 fb = *(const float2*)&sAi[m][k];
            ar[s]  = (v2f){fa.x, fa.y};
            ai[s]  = (v2f){fb.x, fb.y};
            arn[s] = (v2f){-fa.x, -fa.y};
        }

        float bestV[8];
        int   bestC[8];
#pragma unroll
        for (int r = 0; r < 8; ++r) { bestV[r] = -1.0f; bestC[r] = 0; }

        // ---- sweep D: 16 column-tiles per wave, WMMA scoring ----
        for (int t = 0; t < 16; ++t) {
            int ct = wave + 16 * t;
            v8f accR = {}, accI = {};
            const float2* pr = (const float2*)Wr + (size_t)ct * 16 * 32 + lane;
            const float2* pi = (const float2*)Wi + (size_t)ct * 16 * 32 + lane;
#pragma unroll
            for (int s = 0; s < 16; ++s) {
                float2 fr = pr[s * 32];
                float2 fi = pi[s * 32];
                v2f br = (v2f){fr.x, fr.y};
                v2f bi = (v2f){fi.x, fi.y};
                // Re += Ar*Wr + Ai*Wi ; Im += Ai*Wr + (-Ar)*Wi
                accR = __builtin_amdgcn_wmma_f32_16x16x4_f32(
                    false, ar[s],  false, br, (short)0, accR, false, false);
                accR = __builtin_amdgcn_wmma_f32_16x16x4_f32(
                    false, ai[s],  false, bi, (short)0, accR, false, false);
                accI = __builtin_amdgcn_wmma_f32_16x16x4_f32(
                    false, ai[s],  false, br, (short)0, accI, false, false);
                accI = __builtin_amdgcn_wmma_f32_16x16x4_f32(
                    false, arn[s], false, bi, (short)0, accI, false, false);
            }
            // C layout: VGPR r -> row r (lanes 0-15) / row r+8 (lanes 16-31),
            // lane%16 -> column within tile.
            int col = ct * 16 + (lane & 15);
#pragma unroll
            for (int r = 0; r < 8; ++r) {
                float mag = accR[r] * accR[r] + accI[r] * accI[r];
                if (mag > bestV[r]) { bestV[r] = mag; bestC[r] = col; }
            }
        }

        // ---- argmax reduce across the 16 lanes of each half-wave ----
#pragma unroll
        for (int r = 0; r < 8; ++r) {
            for (int msk = 8; msk >= 1; msk >>= 1) {
                float ov = __shfl_xor(bestV[r], msk, 32);
                int   oc = __shfl_xor(bestC[r], msk, 32);
                if (ov > bestV[r]) { bestV[r] = ov; bestC[r] = oc; }
            }
        }
        if (lane == 0) {
#pragma unroll
            for (int r = 0; r < 8; ++r) { redV[wave][r] = bestV[r]; redC[wave][r] = bestC[r]; }
        }
        if (lane == 16) {
#pragma unroll
            for (int r = 0; r < 8; ++r) { redV[wave][r + 8] = bestV[r]; redC[wave][r + 8] = bestC[r]; }
        }
        __syncthreads();

        // ---- cross-wave argmax + exact FP64 coefficient (one thread/row) ----
        if (tid < 16) {
            float bv = -1.0f; int bc = 0;
            for (int w = 0; w < 16; ++w)
                if (redV[w][tid] > bv) { bv = redV[w][tid]; bc = redC[w][tid]; }
            sCol[tid] = bc;
            double sr = 0.0, si = 0.0;
            for (int n = 0; n < NN; ++n) {
                double rr = sRe[tid][n], ri = sIm[tid][n];
                size_t wi_ = (size_t)(n * DD + bc) * 2;
                double wr = W[wi_], wim = W[wi_ + 1];
                // residual . conj(W):  (rr+i ri)(wr - i wim)
                sr += rr * wr + ri * wim;
                si += ri * wr - rr * wim;
            }
            sSelRe[tid] = sr; sSelIm[tid] = si;
        }
        __syncthreads();

        // ---- FP64 rank-1 residual update + refresh FP32 shadow ----
        for (int e = tid; e < 16 * NN; e += 512) {
            int r = e >> 6, n = e & 63;
            int col = sCol[r];
            size_t wi_ = (size_t)(n * DD + col) * 2;
            double wr = W[wi_], wim = W[wi_ + 1];
            double sr = sSelRe[r], si = sSelIm[r];
            double dre = sr * wr - si * wim;
            double dim = sr * wim + si * wr;
            double nr = sRe[r][n] - dre;
            double ni = sIm[r][n] - dim;
            sRe[r][n] = nr;        sIm[r][n] = ni;
            sAr[r][n] = (float)nr; sAi[r][n] = (float)ni;
        }
        __syncthreads();
    }

    // ---- emit residual and x_hat = x - residual ----
    for (int e = tid; e < 16 * NN; e += 512) {
        int r = e >> 6, n = e & 63;
        size_t gi = ((size_t)(row0 + r) * NN + n) * 2;
        double rr = sRe[r][n], ri = sIm[r][n];
        out[gi]     = rr;
        out[gi + 1] = ri;
        out[OUT_HALF + gi]     = x[gi]     - rr;
        out[OUT_HALF + gi + 1] = x[gi + 1] - ri;
    }
}

extern "C" void kernel_launch(void* const* d_in, const int* in_sizes, int n_in,
                              void* d_out, int out_size, void* d_ws, size_t ws_size,
                              hipStream_t stream) {
    const double* x   = (const double*)d_in[0];  // complex128 [2048][64] interleaved
    const double* W   = (const double*)d_in[1];  // complex128 [64][4096] interleaved
    const int*    kp  = (const int*)d_in[2];     // scalar k (=8)
    double*       out = (double*)d_out;

    // Scratch: two packed FP32 views of W (1 MB each).
    float* Wr = (float*)d_ws;
    float* Wi = Wr + (size_t)NN * DD;

    // Pack W once (L2-resident thereafter).
    pack_W_kernel<<<(NN * DD + 255) / 256, 256, 0, stream>>>(W, Wr, Wi);

    // 128 blocks of 16 rows; 16 waves per block.
    mp_wmma_kernel<<<BB / 16, 512, 0, stream>>>(x, W, kp, out, Wr, Wi);
}